// TFLoudnessLoss_13486197309546
// MI455X (gfx1250) — compile-verified
//
#include <hip/hip_runtime.h>
#include <math.h>

typedef float v2f __attribute__((ext_vector_type(2)));
typedef float v8f __attribute__((ext_vector_type(8)));

#define LTOT 160000
#define NB1  256
#define NB2  625
#define PI_F 3.14159265358979f

// ---------------------------------------------------------------------------
// f32 WMMA 16x16x4:  D = A(16x4) * B(4x16) + C(16x16)
// A layout: lane = M + 16*(K>=2), vgpr = K&1.  B mirrors with N over lanes.
// C/D layout: lanes hold N (0..15 twice), vgpr r holds M = r + 8*(lane>=16).
// ---------------------------------------------------------------------------
__device__ __forceinline__ v8f wmma4(v2f a, v2f b, v8f c) {
  return __builtin_amdgcn_wmma_f32_16x16x4_f32(false, a, false, b, (short)0, c,
                                               false, false);
}

// Band id. Exact-rounding analysis of numpy's
//   searchsorted(linspace(0,8000,9), |fftfreq(L,1/SR)|, 'left') - 1  (mod 8)
// shows 160000*(1/16000) rounds to exactly 10.0, val = 0.1, and m*0.1 rounds
// to exactly j*1000.0 at every boundary m = j*10000 (strict '<' fails), so:
__device__ __forceinline__ int band_of(int k) {
  int m = (k < LTOT / 2) ? k : (LTOT - k);   // |signed bin index|
  return (m == 0) ? 7 : ((m + 9999) / 10000 - 1);
}

// ---------------------------------------------------------------------------
// One wave: 256-point DFT (radix-16 x radix-16 via WMMA) + four-step twiddle.
// Input y[n1] in sRe/sIm[256]; output T'[k1] -> outT[k1*625 + n2].
// All twiddle args are integer-reduced to [-2pi, 2pi] so the fast hardware
// __sincosf (v_sin_f32/v_cos_f32) is used -- no Payne-Hanek slow path.
// ---------------------------------------------------------------------------
__device__ __forceinline__ void dft256_wmma(float* sRe, float* sIm, int lane,
                                            int n2, float2* __restrict__ outT) {
  const int lm    = lane & 15;          // M for A / N for B / N for C-D
  const int khalf = (lane >> 4) << 1;   // 0 or 2 (K sub-offset)
  const int moff  = (lane >> 4) << 3;   // 0 or 8 (C/D row base)

  // ---- GEMM1: T = D16 * X,  X[a][b] = y[16a+b] ----
  v8f tre = {}; v8f tim = {};
#pragma unroll
  for (int kc = 0; kc < 4; ++kc) {
    const int K0 = 4 * kc + khalf;
    float c0, s0, c1, s1;
    __sincosf((2.0f * PI_F / 16.0f) * (float)((lm * K0) & 15), &s0, &c0);
    __sincosf((2.0f * PI_F / 16.0f) * (float)((lm * (K0 + 1)) & 15), &s1, &c1);
    v2f Ac = {c0, c1};                  // cos(2pi*M*K/16)
    v2f As = {s0, s1};                  // sin(2pi*M*K/16)
    v2f Bre = {sRe[16 * K0 + lm], sRe[16 * (K0 + 1) + lm]};
    v2f Bim = {sIm[16 * K0 + lm], sIm[16 * (K0 + 1) + lm]};
    v2f BreN = {-Bre.x, -Bre.y};
    // D16 = C - iS  =>  Tre = C*Xre + S*Xim ;  Tim = C*Xim - S*Xre
    tre = wmma4(Ac, Bre, tre);
    tre = wmma4(As, Bim, tre);
    tim = wmma4(Ac, Bim, tim);
    tim = wmma4(As, BreN, tim);
  }
  __syncthreads();                      // done reading X from LDS

  // ---- twiddle w256^{c*b} (c*b <= 225 < 256); store T' at [c*16+b] ----
#pragma unroll
  for (int r = 0; r < 8; ++r) {
    int M = r + moff;                   // c
    int Nn = lm;                        // b
    float ss, cc;
    __sincosf(-(2.0f * PI_F / 256.0f) * (float)(M * Nn), &ss, &cc);
    float xr = tre[r], xi = tim[r];
    sRe[M * 16 + Nn] = xr * cc - xi * ss;
    sIm[M * 16 + Nn] = xr * ss + xi * cc;
  }
  __syncthreads();

  // ---- GEMM2: U = T' * D16 ----
  v8f ure = {}; v8f uim = {};
#pragma unroll
  for (int kc = 0; kc < 4; ++kc) {
    const int K0 = 4 * kc + khalf;
    v2f Are = {sRe[lm * 16 + K0], sRe[lm * 16 + K0 + 1]};
    v2f Aim = {sIm[lm * 16 + K0], sIm[lm * 16 + K0 + 1]};
    v2f AreN = {-Are.x, -Are.y};
    float c0, s0, c1, s1;
    __sincosf((2.0f * PI_F / 16.0f) * (float)((K0 * lm) & 15), &s0, &c0);
    __sincosf((2.0f * PI_F / 16.0f) * (float)(((K0 + 1) * lm) & 15), &s1, &c1);
    v2f Bc = {c0, c1};
    v2f Bs = {s0, s1};
    // Ure = T're*C + T'im*S ;  Uim = T'im*C - T're*S
    ure = wmma4(Are, Bc, ure);
    ure = wmma4(Aim, Bs, ure);
    uim = wmma4(Aim, Bc, uim);
    uim = wmma4(AreN, Bs, uim);
  }

  // ---- four-step twiddle w_L^{n2*k1} (n2*k1 < L); U[c][d] = Y[c+16d] ----
#pragma unroll
  for (int r = 0; r < 8; ++r) {
    int M = r + moff;                   // c
    int Nn = lm;                        // d
    int k1 = M + 16 * Nn;
    float ss, cc;
    __sincosf(-(2.0f * PI_F / (float)LTOT) * (float)(n2 * k1), &ss, &cc);
    float xr = ure[r], xi = uim[r];
    outT[(size_t)k1 * NB2 + n2] = make_float2(xr * cc - xi * ss, xr * ss + xi * cc);
  }
}

// ---- step 1+2 forward: real input, one wave per (transform, n2) ----
__global__ void __launch_bounds__(32)
f1_fwd(const float* __restrict__ wsig, const float* __restrict__ osig,
       float2* __restrict__ Tg) {
  __shared__ float sRe[256], sIm[256];
  int bid = blockIdx.x;
  int tr  = bid / NB2;                  // 0..7 watermarked, 8..15 original
  int n2  = bid % NB2;
  int lane = threadIdx.x;
  const float* src = (tr < 8) ? (wsig + (size_t)tr * LTOT)
                              : (osig + (size_t)(tr - 8) * LTOT);
#pragma unroll
  for (int i = 0; i < 8; ++i) {
    int n1 = lane + 32 * i;
    sRe[n1] = src[(size_t)n1 * NB2 + n2];
    sIm[n1] = 0.0f;
  }
  __syncthreads();
  dft256_wmma(sRe, sIm, lane, n2, Tg + (size_t)tr * LTOT);
}

// ---- step 1+2 inverse (band-masked, conjugated spectrum) ----
__global__ void __launch_bounds__(32)
f1_inv(const float2* __restrict__ spec, float2* __restrict__ Tg, int band) {
  __shared__ float sRe[256], sIm[256];
  int bid = blockIdx.x;
  int tr  = bid / NB2;
  int n2  = bid % NB2;
  int lane = threadIdx.x;
  const float2* S = spec + (size_t)tr * LTOT;
#pragma unroll
  for (int i = 0; i < 8; ++i) {
    int n1 = lane + 32 * i;
    int k  = n1 * NB2 + n2;
    float re = 0.0f, im = 0.0f;
    if (band_of(k) == band) { float2 v = S[k]; re = v.x; im = -v.y; }
    sRe[n1] = re;
    sIm[n1] = im;
  }
  __syncthreads();
  dft256_wmma(sRe, sIm, lane, n2, Tg + (size_t)tr * LTOT);
}

// ---------------------------------------------------------------------------
// 625-point Stockham radix-5 FFT in LDS
// ---------------------------------------------------------------------------
__device__ __forceinline__ void dft5(float2 v[5]) {
  const float WR[5] = {1.0f, 0.309016994f, -0.809016994f, -0.809016994f, 0.309016994f};
  const float WI[5] = {0.0f, -0.951056516f, -0.587785252f, 0.587785252f, 0.951056516f};
  float2 o[5];
#pragma unroll
  for (int k = 0; k < 5; ++k) {
    float ore = 0.0f, oim = 0.0f;
#pragma unroll
    for (int r = 0; r < 5; ++r) {
      int m = (r * k) % 5;
      ore += v[r].x * WR[m] - v[r].y * WI[m];
      oim += v[r].x * WI[m] + v[r].y * WR[m];
    }
    o[k] = make_float2(ore, oim);
  }
#pragma unroll
  for (int k = 0; k < 5; ++k) v[k] = o[k];
}

__device__ __forceinline__ void fft625_block(float2* A, float2* B, int tid) {
  int Ns = 1;
  float2* s = A;
  float2* d = B;
#pragma unroll
  for (int st = 0; st < 4; ++st) {
    if (tid < 125) {
      float2 v[5];
#pragma unroll
      for (int r = 0; r < 5; ++r) v[r] = s[tid + 125 * r];
      int jm = tid % Ns;
      float base = -(2.0f * PI_F) * (float)jm / (float)(5 * Ns);
#pragma unroll
      for (int r = 1; r < 5; ++r) {
        float ss, cc;
        __sincosf(base * (float)r, &ss, &cc);   // |arg| < 2pi
        float xr = v[r].x, xi = v[r].y;
        v[r] = make_float2(xr * cc - xi * ss, xr * ss + xi * cc);
      }
      dft5(v);
      int idxD = (tid / Ns) * (Ns * 5) + jm;
#pragma unroll
      for (int r = 0; r < 5; ++r) d[idxD + Ns * r] = v[r];
    }
    __syncthreads();
    float2* t = s; s = d; d = t;
    Ns *= 5;
  }
  // result resides in A after 4 swaps
}

// ---- step 3 forward: write spectrum X[k1 + 256*k2] ----
__global__ void __launch_bounds__(128)
f2_fwd(const float2* __restrict__ Tg, float2* __restrict__ spec) {
  __shared__ float2 A[625], B[625];
  int bid = blockIdx.x;
  int tr = bid >> 8;
  int k1 = bid & 255;
  int tid = threadIdx.x;
  const float2* in = Tg + (size_t)tr * LTOT + (size_t)k1 * NB2;
  for (int i = tid; i < NB2; i += 128) A[i] = in[i];
  __syncthreads();
  fft625_block(A, B, tid);
  float2* out = spec + (size_t)tr * LTOT;
  for (int i = tid; i < NB2; i += 128) out[k1 + 256 * i] = A[i];
}

// ---- step 3 inverse: accumulate half-window energies of Re(.) into P ----
__global__ void __launch_bounds__(128)
f2_inv(const float2* __restrict__ Tg, float* __restrict__ P, int band) {
  __shared__ float2 A[625], B[625];
  __shared__ float Pl[157];
  int bid = blockIdx.x;
  int tr = bid >> 8;
  int k1 = bid & 255;
  int tid = threadIdx.x;
  const float2* in = Tg + (size_t)tr * LTOT + (size_t)k1 * NB2;
  for (int i = tid; i < NB2; i += 128) A[i] = in[i];
  for (int i = tid; i < 157; i += 128) Pl[i] = 0.0f;
  __syncthreads();
  fft625_block(A, B, tid);
  for (int i = tid; i < NB2; i += 128) {
    int t = k1 + 256 * i;                 // time index in [0, 160000)
    float re = A[i].x;                    // un-scaled Re(ifft)*L
    atomicAdd(&Pl[t >> 10], re * re);
  }
  __syncthreads();
  float* Pt = P + ((size_t)tr * 8 + band) * 156;
  for (int i = tid; i < 156; i += 128) atomicAdd(&Pt[i], Pl[i]);
}

// ---------------------------------------------------------------------------
// Final: loudness diffs (constants cancel), softmax-weighted sum -> scalar
// ---------------------------------------------------------------------------
__global__ void __launch_bounds__(1024)
loss_kernel(const float* __restrict__ P, float* __restrict__ out) {
  __shared__ float dbuf[9920];
  __shared__ float red[1024];
  int tid = threadIdx.x;
  const float invDen = 1.0f / (2048.0f * (float)LTOT * (float)LTOT);
  float lmax = -3.4e38f;
  for (int i = tid; i < 9920; i += 1024) {
    int b = i / (8 * 155);
    int band = (i / 155) % 8;
    int j = i % 155;
    const float* Pw = P + ((size_t)b * 8 + band) * 156;        // watermarked
    const float* Po = P + ((size_t)(8 + b) * 8 + band) * 156;  // original
    float msw = (Pw[j] + Pw[j + 1]) * invDen + 1e-12f;
    float mso = (Po[j] + Po[j + 1]) * invDen + 1e-12f;
    float dv = 10.0f * (__log10f(msw) - __log10f(mso));
    dbuf[i] = dv;
    lmax = fmaxf(lmax, dv);
  }
  red[tid] = lmax;
  __syncthreads();
  for (int s = 512; s > 0; s >>= 1) {
    if (tid < s) red[tid] = fmaxf(red[tid], red[tid + s]);
    __syncthreads();
  }
  float m = red[0];
  __syncthreads();
  float s1 = 0.0f, s2 = 0.0f;
  for (int i = tid; i < 9920; i += 1024) {
    float dv = dbuf[i];
    float e = __expf(dv - m);
    s1 += e;
    s2 += e * dv;
  }
  red[tid] = s1;
  __syncthreads();
  for (int s = 512; s > 0; s >>= 1) {
    if (tid < s) red[tid] += red[tid + s];
    __syncthreads();
  }
  float S1 = red[0];
  __syncthreads();
  red[tid] = s2;
  __syncthreads();
  for (int s = 512; s > 0; s >>= 1) {
    if (tid < s) red[tid] += red[tid + s];
    __syncthreads();
  }
  if (tid == 0) out[0] = red[0] / S1;
}

// ---------------------------------------------------------------------------
extern "C" void kernel_launch(void* const* d_in, const int* in_sizes, int n_in,
                              void* d_out, int out_size, void* d_ws,
                              size_t ws_size, hipStream_t stream) {
  const float* wsig = (const float*)d_in[0];   // [8][160000] f32
  const float* osig = (const float*)d_in[1];   // [8][160000] f32

  char* base = (char*)d_ws;
  float2* Tg   = (float2*)base;                                        // 20.48 MB
  float2* spec = (float2*)(base + (size_t)16 * LTOT * sizeof(float2)); // 20.48 MB
  float*  P    = (float*)(base + (size_t)32 * LTOT * sizeof(float2));  // 80 KB

  hipMemsetAsync(P, 0, (size_t)16 * 8 * 156 * sizeof(float), stream);

  // forward FFTs of all 16 signals
  f1_fwd<<<16 * NB2, 32, 0, stream>>>(wsig, osig, Tg);
  f2_fwd<<<16 * 256, 128, 0, stream>>>(Tg, spec);

  // per-band masked inverse FFTs, fused energy accumulation
  for (int band = 0; band < 8; ++band) {
    f1_inv<<<16 * NB2, 32, 0, stream>>>(spec, Tg, band);
    f2_inv<<<16 * 256, 128, 0, stream>>>(Tg, P, band);
  }

  loss_kernel<<<1, 1024, 0, stream>>>(P, (float*)d_out);
}